// AlgoMiniBatch_82059645157376
// MI455X (gfx1250) — compile-verified
//
#include <hip/hip_runtime.h>
#include <hip/hip_bf16.h>

typedef __attribute__((ext_vector_type(2))) float v2f;
typedef __attribute__((ext_vector_type(8))) float v8f;

#define NB        1024      // batch (n2)
#define DD        256       // feature dim
#define KK        512       // concat dim = 2*D
#define S0        25        // fanout layer 0
#define S1        10        // fanout layer 1
#define N1        (NB * (1 + S1))   // 11264 rows at layer 0 output
#define TILE_K    128

// ---------------- zero scratch (rowsq accumulators) ----------------
__global__ void k_zero(float* __restrict__ p, int n) {
    int i = blockIdx.x * blockDim.x + threadIdx.x;
    if (i < n) p[i] = 0.0f;
}

// ---------------- layer-0 gather + mean aggregate ----------------
// x0[i, 0:256]   = features[nodes1[i]]
// x0[i, 256:512] = mean_s features[neigh1[i,s]]
__global__ __launch_bounds__(256) void k_agg0(
    const float* __restrict__ feats, const int* __restrict__ nodes2,
    const int* __restrict__ neigh2, const int* __restrict__ neigh1,
    float* __restrict__ x0)
{
    const int i = blockIdx.x;          // row in [0, N1)
    const int t = threadIdx.x;         // column in [0, 256)
    const int self = (i < NB) ? nodes2[i] : neigh2[i - NB];
    x0[(size_t)i * KK + t] = feats[(size_t)self * DD + t];
    float acc = 0.0f;
    const int* nb = &neigh1[(size_t)i * S0];
    #pragma unroll 5
    for (int s = 0; s < S0; ++s)
        acc += feats[(size_t)nb[s] * DD + t];
    x0[(size_t)i * KK + DD + t] = acc / (float)S0;
}

// ---------------- layer-1 gather + mean aggregate ----------------
__global__ __launch_bounds__(256) void k_agg1(
    const float* __restrict__ h1, float* __restrict__ x1)
{
    const int i = blockIdx.x;          // row in [0, NB)
    const int t = threadIdx.x;
    x1[(size_t)i * KK + t] = h1[(size_t)i * DD + t];
    float acc = 0.0f;
    #pragma unroll
    for (int s = 0; s < S1; ++s)
        acc += h1[(size_t)(NB + i * S1 + s) * DD + t];
    x1[(size_t)i * KK + DD + t] = acc / (float)S1;
}

// ---------------- fp32 WMMA GEMM: Y = relu(X @ W^T + bias), rowsq += sum(Y^2) ----
// X: (nrows, 512), W: (256, 512) row-major, Y: (nrows, 256)
// grid = (nrows/16, 256/64), block = 128 threads (4 waves), wave w -> col tile w
__global__ __launch_bounds__(128) void k_gemm_relu_sq(
    const float* __restrict__ X, const float* __restrict__ W,
    const float* __restrict__ bias, float* __restrict__ Y,
    float* __restrict__ rowsq)
{
    __shared__ float Xs[16][TILE_K + 4];   // +4 pad: conflict-free b64 reads
    __shared__ float Ws[64][TILE_K + 4];

    const int rb   = blockIdx.x * 16;      // row tile base
    const int cb   = blockIdx.y * 64;      // col group base
    const int lane = threadIdx.x & 31;
    const int wave = threadIdx.x >> 5;
    const int mrow = lane & 15;            // M for A-frag, N for B-frag
    const int koff = (lane >> 4) << 1;     // 0 or 2 (K sub-offset per ISA layout)

    v8f c = {};

    for (int kc = 0; kc < KK; kc += TILE_K) {
        // stage W tile: 64 rows x TILE_K, coalesced float4
        for (int f = threadIdx.x; f < 64 * (TILE_K / 4); f += 128) {
            int r  = f >> 5;
            int c4 = (f & 31) << 2;
            *(float4*)&Ws[r][c4] =
                *(const float4*)&W[(size_t)(cb + r) * KK + kc + c4];
        }
        // stage X tile: 16 rows x TILE_K
        for (int f = threadIdx.x; f < 16 * (TILE_K / 4); f += 128) {
            int r  = f >> 5;
            int c4 = (f & 31) << 2;
            *(float4*)&Xs[r][c4] =
                *(const float4*)&X[(size_t)(rb + r) * KK + kc + c4];
        }
        __syncthreads();

        #pragma unroll
        for (int k = 0; k < TILE_K; k += 4) {
            v2f a = *(const v2f*)&Xs[mrow][k + koff];
            v2f b = *(const v2f*)&Ws[wave * 16 + mrow][k + koff];
            // D = A(16x4 f32) x B(4x16 f32) + C(16x16 f32)
            c = __builtin_amdgcn_wmma_f32_16x16x4_f32(
                    false, a, false, b, (short)0, c, false, false);
        }
        __syncthreads();
    }

    // epilogue: bias + relu + store + row sum-of-squares
    const int n    = cb + wave * 16 + mrow;
    const float bn = bias[n];
    const int rsel = (lane >> 4) << 3;     // lanes 16-31 hold M = r + 8
    #pragma unroll
    for (int r = 0; r < 8; ++r) {
        const int row = rb + rsel + r;
        float v = c[r] + bn;
        v = v > 0.0f ? v : 0.0f;
        Y[(size_t)row * DD + n] = v;
        float sq = v * v;                  // reduce over the 16 N-lanes
        sq += __shfl_xor(sq, 1, 32);
        sq += __shfl_xor(sq, 2, 32);
        sq += __shfl_xor(sq, 4, 32);
        sq += __shfl_xor(sq, 8, 32);
        if (mrow == 0) atomicAdd(&rowsq[row], sq);
    }
}

// ---------------- row L2 normalize (norm==0 -> scale 1) ----------------
__global__ __launch_bounds__(256) void k_norm(
    const float* __restrict__ Y, const float* __restrict__ rowsq,
    float* __restrict__ out)
{
    const int i = blockIdx.x;
    const int t = threadIdx.x;
    const float ss = rowsq[i];
    const float scale = (ss > 0.0f) ? (1.0f / sqrtf(ss)) : 1.0f;
    out[(size_t)i * DD + t] = Y[(size_t)i * DD + t] * scale;
}

extern "C" void kernel_launch(void* const* d_in, const int* in_sizes, int n_in,
                              void* d_out, int out_size, void* d_ws, size_t ws_size,
                              hipStream_t stream) {
    const float* feats  = (const float*)d_in[0];   // (100000, 256)
    const float* W0     = (const float*)d_in[1];   // (256, 512)
    const float* b0     = (const float*)d_in[2];   // (256,)
    const float* W1     = (const float*)d_in[3];   // (256, 512)
    const float* b1     = (const float*)d_in[4];   // (256,)
    const int*   nodes2 = (const int*)d_in[5];     // (1024,)
    const int*   neigh2 = (const int*)d_in[6];     // (1024, 10) flat
    const int*   neigh1 = (const int*)d_in[7];     // (11264, 25) flat
    float* out = (float*)d_out;                    // (1024, 256)

    // workspace layout (floats)
    float* ws = (float*)d_ws;
    size_t o = 0;
    float* x0     = ws + o; o += (size_t)N1 * KK;   // 5,767,168
    float* Y1     = ws + o; o += (size_t)N1 * DD;   // 2,883,584
    float* h1     = ws + o; o += (size_t)N1 * DD;   // 2,883,584
    float* x1     = ws + o; o += (size_t)NB * KK;   //   524,288
    float* Y2     = ws + o; o += (size_t)NB * DD;   //   262,144
    float* rowsq1 = ws + o; o += N1;                //    11,264
    float* rowsq2 = ws + o; o += NB;                //     1,024
    (void)ws_size;

    // 0) zero both rowsq accumulators (rowsq1 and rowsq2 are contiguous)
    {
        int n = N1 + NB;
        k_zero<<<(n + 255) / 256, 256, 0, stream>>>(rowsq1, n);
    }
    // 1) layer-0 gather/aggregate -> x0 (11264, 512)
    k_agg0<<<N1, 256, 0, stream>>>(feats, nodes2, neigh2, neigh1, x0);
    // 2) h1_pre = relu(x0 @ W0^T + b0), rowsq1
    {
        dim3 grid(N1 / 16, DD / 64);
        k_gemm_relu_sq<<<grid, 128, 0, stream>>>(x0, W0, b0, Y1, rowsq1);
    }
    // 3) h1 = l2_normalize(h1_pre)
    k_norm<<<N1, 256, 0, stream>>>(Y1, rowsq1, h1);
    // 4) layer-1 gather/aggregate -> x1 (1024, 512)
    k_agg1<<<NB, 256, 0, stream>>>(h1, x1);
    // 5) h2_pre = relu(x1 @ W1^T + b1), rowsq2
    {
        dim3 grid(NB / 16, DD / 64);
        k_gemm_relu_sq<<<grid, 128, 0, stream>>>(x1, W1, b1, Y2, rowsq2);
    }
    // 6) output = l2_normalize(h2_pre)
    k_norm<<<NB, 256, 0, stream>>>(Y2, rowsq2, out);
}